// TTTLinearLayer_29875792511772
// MI455X (gfx1250) — compile-verified
//
#include <hip/hip_runtime.h>

#define TTT_LR 0.01f
#define LN_EPS 1e-5f

// Problem constants (from reference setup_inputs)
#define BB 4
#define SS 2048
#define DD 1024
#define DS 64
#define CHUNK 64

typedef float v2f __attribute__((ext_vector_type(2)));
typedef float v8f __attribute__((ext_vector_type(8)));

// ---------------- WMMA f32 16x16x4 fragment helpers (wave32) ----------------
// A (16x4, MxK): lanes 0-15 -> M=lane, K = k0+{0,1}; lanes 16-31 -> K = k0+{2,3}
__device__ __forceinline__ v2f ld_a(const float* p, int ld, int m0, int k0, int lane) {
  int m = m0 + (lane & 15);
  int k = k0 + ((lane >> 4) << 1);
  v2f a; a.x = p[m * ld + k]; a.y = p[m * ld + k + 1]; return a;
}
// B (4x16, KxN) from row-major p[k][n]
__device__ __forceinline__ v2f ld_b(const float* p, int ld, int k0, int n0, int lane) {
  int n = n0 + (lane & 15);
  int k = k0 + ((lane >> 4) << 1);
  v2f b; b.x = p[k * ld + n]; b.y = p[(k + 1) * ld + n]; return b;
}
// B (4x16, KxN) from transposed view: B[k][n] = p[n*ld + k]
__device__ __forceinline__ v2f ld_bt(const float* p, int ld, int k0, int n0, int lane) {
  int n = n0 + (lane & 15);
  int k = k0 + ((lane >> 4) << 1);
  v2f b; b.x = p[n * ld + k]; b.y = p[n * ld + k + 1]; return b;
}
// C/D (16x16): vgpr r -> M = m0 + r + 8*(lane>=16), N = n0 + (lane&15)
__device__ __forceinline__ v8f ld_c(const float* p, int ld, int m0, int n0, int lane) {
  int n = n0 + (lane & 15);
  int mb = m0 + ((lane >> 4) << 3);
  v8f c;
#pragma unroll
  for (int r = 0; r < 8; ++r) c[r] = p[(mb + r) * ld + n];
  return c;
}
__device__ __forceinline__ void st_c(float* p, int ld, int m0, int n0, int lane, v8f c) {
  int n = n0 + (lane & 15);
  int mb = m0 + ((lane >> 4) << 3);
#pragma unroll
  for (int r = 0; r < 8; ++r) p[(mb + r) * ld + n] = c[r];
}

#define WMMA_F32(a, b, c) \
  __builtin_amdgcn_wmma_f32_16x16x4_f32(false, (a), false, (b), (short)0, (c), false, false)

// ---------------- Kernel 1: LayerNorm (row = one block, 256 threads) --------
__global__ void __launch_bounds__(256) ln_kernel(
    const float* __restrict__ x, const float* __restrict__ gamma,
    const float* __restrict__ beta, float* __restrict__ h) {
  __shared__ float ssum[256];
  __shared__ float ssq[256];
  const int row = blockIdx.x;
  const int t = threadIdx.x;
  const float4 v = ((const float4*)(x + (size_t)row * DD))[t];
  float s = v.x + v.y + v.z + v.w;
  float q = v.x * v.x + v.y * v.y + v.z * v.z + v.w * v.w;
  ssum[t] = s; ssq[t] = q;
  __syncthreads();
  for (int off = 128; off > 0; off >>= 1) {
    if (t < off) { ssum[t] += ssum[t + off]; ssq[t] += ssq[t + off]; }
    __syncthreads();
  }
  const float mu = ssum[0] * (1.0f / DD);
  const float var = ssq[0] * (1.0f / DD) - mu * mu;
  const float rs = rsqrtf(var + LN_EPS);
  const float4 g = ((const float4*)gamma)[t];
  const float4 b = ((const float4*)beta)[t];
  float4 o;
  o.x = (v.x - mu) * rs * g.x + b.x;
  o.y = (v.y - mu) * rs * g.y + b.y;
  o.z = (v.z - mu) * rs * g.z + b.z;
  o.w = (v.w - mu) * rs * g.w + b.w;
  ((float4*)(h + (size_t)row * DD))[t] = o;
}

// ---------------- Kernel 2: projection GEMM  C[M x N] = H[M x K] W[N x K]^T + b
// block = 256 threads (8 waves), block tile 64x64, K = 1024.
// Software-pipelined: next k-tile is loaded global->regs while WMMAs run on
// the current LDS tile; s_wait_loadcnt lands at the reg->LDS store.
__global__ void __launch_bounds__(256) proj_kernel(
    const float* __restrict__ H, const float* __restrict__ W,
    const float* __restrict__ bias, float* __restrict__ C, int N) {
  __shared__ float sH[64 * 65];
  __shared__ float sB[64 * 65];
  const int tid = threadIdx.x;
  const int lane = tid & 31;
  const int wv = tid >> 5;
  const int m0g = blockIdx.x * 64;
  const int n0g = blockIdx.y * 64;
  const int tr = (wv >> 1) * 16;
  const int tc0 = (wv & 1) * 32;
  v8f acc0 = 0.0f, acc1 = 0.0f;

  // prologue: stage k-tile 0
#pragma unroll
  for (int rep = 0; rep < 16; ++rep) {
    int e = rep * 256 + tid;
    int r = e >> 6, cx = e & 63;
    sH[r * 65 + cx] = H[(size_t)(m0g + r) * DD + cx];
    sB[r * 65 + cx] = W[(size_t)(n0g + r) * DD + cx];
  }
  __syncthreads();

  for (int kt = 0; kt < DD; kt += 64) {
    float hreg[16], wreg[16];
    const bool more = (kt + 64) < DD;
    if (more) {
#pragma unroll
      for (int rep = 0; rep < 16; ++rep) {
        int e = rep * 256 + tid;
        int r = e >> 6, cx = e & 63;
        hreg[rep] = H[(size_t)(m0g + r) * DD + kt + 64 + cx];
        wreg[rep] = W[(size_t)(n0g + r) * DD + kt + 64 + cx];
      }
    }
#pragma unroll
    for (int k0 = 0; k0 < 64; k0 += 4) {
      v2f a  = ld_a(sH, 65, tr, k0, lane);
      v2f b0 = ld_bt(sB, 65, k0, tc0, lane);
      v2f b1 = ld_bt(sB, 65, k0, tc0 + 16, lane);
      acc0 = WMMA_F32(a, b0, acc0);
      acc1 = WMMA_F32(a, b1, acc1);
    }
    __syncthreads();
    if (more) {
#pragma unroll
      for (int rep = 0; rep < 16; ++rep) {
        int e = rep * 256 + tid;
        int r = e >> 6, cx = e & 63;
        sH[r * 65 + cx] = hreg[rep];
        sB[r * 65 + cx] = wreg[rep];
      }
    }
    __syncthreads();
  }
  const int nA = n0g + tc0 + (lane & 15);
  const int mb = m0g + tr + ((lane >> 4) << 3);
  const float bz0 = bias[nA];
  const float bz1 = bias[nA + 16];
#pragma unroll
  for (int r = 0; r < 8; ++r) {
    C[(size_t)(mb + r) * N + nA]      = acc0[r] + bz0;
    C[(size_t)(mb + r) * N + nA + 16] = acc1[r] + bz1;
  }
}

// ---------------- Kernel 3: chunked TTT scan --------------------------------
// grid = (B, D/128). Each block owns a 128x64 slice of the fast-weight W for
// one batch and walks 32 chunks of 64 timesteps. All dense work is fp32 WMMA.
// The 32KB V chunk is DMA'd into LDS (global_load_async_to_lds_b128, ASYNCcnt)
// at chunk start, overlapping the K/Q loads and the A/P GEMMs; next-chunk
// K/Q/x streams are prefetched into L2 (global_prefetch_b8).
__global__ void __launch_bounds__(256, 1) scan_kernel(
    const float* __restrict__ x, const float* __restrict__ keys,
    const float* __restrict__ queries, const float* __restrict__ values,
    float* __restrict__ out) {
  __shared__ float sW[128 * 65];   // W slice  [d_local][s]
  __shared__ float sK[64 * 65];    // K chunk  [t][s]
  __shared__ float sQ[64 * 65];    // Q chunk  [t][s]
  __shared__ float sA[64 * 65];    // K K^T, later strict-lower(Q K^T)*(-lr)
  __shared__ float sE[64 * 129];   // P -> R -> E  [t][d_local]
  __shared__ float sY[64 * 129];   // V stage (unpadded 64x128 view), then Y
  const int tid = threadIdx.x;
  const int lane = tid & 31;
  const int wv = tid >> 5;
  const int b = blockIdx.x;
  const int d0 = blockIdx.y * 128;

  for (int i = tid; i < 128 * 65; i += 256) sW[i] = 0.0f;
  __syncthreads();

  const unsigned sy_lds = (unsigned)(uintptr_t)(&sY[0]);  // LDS byte offset

  for (int c = 0; c < SS / CHUNK; ++c) {
    const size_t rowbase = (size_t)b * SS + (size_t)c * CHUNK;
    const float* gk = keys + rowbase * DS;
    const float* gq = queries + rowbase * DS;
    const float* gv = values + rowbase * DD + d0;
    const float* gx = x + rowbase * DD + d0;
    float* go = out + rowbase * DD + d0;

    // ---- async DMA: V chunk (64x128 fp32) global -> LDS staging (in sY) ----
    // 2048 b128 transfers; per lane: VDST = LDS byte addr, VADDR = byte offset,
    // SADDR = chunk base. Tracked on ASYNCcnt.
#pragma unroll
    for (int rep = 0; rep < 8; ++rep) {
      int e = rep * 256 + tid;            // float4 index, 32 per row
      int t = e >> 5, d4 = e & 31;
      unsigned ldsoff = sy_lds + (unsigned)(t * 128 + d4 * 4) * 4u;
      unsigned goff = (unsigned)(t * DD + d4 * 4) * 4u;
      asm volatile("global_load_async_to_lds_b128 %0, %1, %2"
                   :: "v"(ldsoff), "v"(goff), "s"(gv) : "memory");
    }

    // ---- L2 prefetch of next chunk's K/Q/x streams ----
    if (c + 1 < SS / CHUNK) {
      const float* nk = keys + (rowbase + CHUNK) * DS;
      const float* nq = queries + (rowbase + CHUNK) * DS;
      const float* nx = x + (rowbase + CHUNK) * DD + d0;
      __builtin_prefetch(nk + tid * 16, 0, 0);   // 16KB, 64B/thread
      __builtin_prefetch(nq + tid * 16, 0, 0);
      __builtin_prefetch(nx + tid * 32, 0, 0);   // 32KB, 128B/thread
    }

    // ---- load K, Q chunks (coalesced) ----
#pragma unroll
    for (int rep = 0; rep < 16; ++rep) {
      int e = rep * 256 + tid;
      int t = e >> 6, s = e & 63;
      sK[t * 65 + s] = gk[t * DS + s];
      sQ[t * 65 + s] = gq[t * DS + s];
    }
    __syncthreads();

    // ---- A = K K^T (2 tiles/wave)  and  P = K W^T (4 tiles/wave) ----
    {
      const int t0 = wv * 2;
      const int tr0 = (t0 >> 2) * 16, tc0 = (t0 & 3) * 16;
      const int tr1 = ((t0 + 1) >> 2) * 16, tc1 = ((t0 + 1) & 3) * 16;
      v8f a0 = 0.0f, a1 = 0.0f;
      const int trP = (wv >> 1) * 16;
      v8f p0 = 0.0f, p1 = 0.0f, p2 = 0.0f, p3 = 0.0f;
#pragma unroll
      for (int k0 = 0; k0 < 64; k0 += 4) {
        v2f ka0 = ld_a(sK, 65, tr0, k0, lane);
        a0 = WMMA_F32(ka0, ld_bt(sK, 65, k0, tc0, lane), a0);
        v2f ka1 = ld_a(sK, 65, tr1, k0, lane);
        a1 = WMMA_F32(ka1, ld_bt(sK, 65, k0, tc1, lane), a1);
        v2f pa = ld_a(sK, 65, trP, k0, lane);
        const int cb = (wv & 1) * 4;
        p0 = WMMA_F32(pa, ld_bt(sW, 65, k0, (cb + 0) * 16, lane), p0);
        p1 = WMMA_F32(pa, ld_bt(sW, 65, k0, (cb + 1) * 16, lane), p1);
        p2 = WMMA_F32(pa, ld_bt(sW, 65, k0, (cb + 2) * 16, lane), p2);
        p3 = WMMA_F32(pa, ld_bt(sW, 65, k0, (cb + 3) * 16, lane), p3);
      }
      st_c(sA, 65, tr0, tc0, lane, a0);
      st_c(sA, 65, tr1, tc1, lane, a1);
      const int cb = (wv & 1) * 4;
      st_c(sE, 129, trP, (cb + 0) * 16, lane, p0);
      st_c(sE, 129, trP, (cb + 1) * 16, lane, p1);
      st_c(sE, 129, trP, (cb + 2) * 16, lane, p2);
      st_c(sE, 129, trP, (cb + 3) * 16, lane, p3);
    }

    // ---- R = P - V : wait for this wave's DMA, barrier, then subtract ----
    asm volatile("s_wait_asynccnt 0" ::: "memory");
    __syncthreads();
#pragma unroll
    for (int rep = 0; rep < 32; ++rep) {
      int e = rep * 256 + tid;
      int t = e >> 7, d = e & 127;
      sE[t * 129 + d] -= sY[t * 128 + d];   // unpadded staging view
    }
    __syncthreads();

    // ---- waves 0-3: forward substitution  E = (I + lr*stril(A))^-1 R
    //      (column-private: no barriers needed inside)
    //      waves 4-7: Y = Q W^T via WMMA (independent buffers) ----
    if (wv < 4) {
      const int d = tid;  // 0..127, one LDS column per thread
      for (int i = 1; i < 64; ++i) {
        float acc = sE[i * 129 + d];
        for (int t = 0; t < i; ++t)
          acc -= TTT_LR * sA[i * 65 + t] * sE[t * 129 + d];
        sE[i * 129 + d] = acc;
      }
    } else {
      const int trY = (wv - 4) * 16;
      v8f y[8];
#pragma unroll
      for (int i = 0; i < 8; ++i) y[i] = 0.0f;
#pragma unroll
      for (int k0 = 0; k0 < 64; k0 += 4) {
        v2f qa = ld_a(sQ, 65, trY, k0, lane);
#pragma unroll
        for (int i = 0; i < 8; ++i)
          y[i] = WMMA_F32(qa, ld_bt(sW, 65, k0, i * 16, lane), y[i]);
      }
#pragma unroll
      for (int i = 0; i < 8; ++i) st_c(sY, 129, trY, i * 16, lane, y[i]);
    }
    __syncthreads();

    // ---- Bq = Q K^T  -> sA (A no longer needed) ----
    {
      const int t0 = wv * 2;
      const int tr0 = (t0 >> 2) * 16, tc0 = (t0 & 3) * 16;
      const int tr1 = ((t0 + 1) >> 2) * 16, tc1 = ((t0 + 1) & 3) * 16;
      v8f a0 = 0.0f, a1 = 0.0f;
#pragma unroll
      for (int k0 = 0; k0 < 64; k0 += 4) {
        a0 = WMMA_F32(ld_a(sQ, 65, tr0, k0, lane), ld_bt(sK, 65, k0, tc0, lane), a0);
        a1 = WMMA_F32(ld_a(sQ, 65, tr1, k0, lane), ld_bt(sK, 65, k0, tc1, lane), a1);
      }
      st_c(sA, 65, tr0, tc0, lane, a0);
      st_c(sA, 65, tr1, tc1, lane, a1);
    }
    __syncthreads();

    // ---- mask: sA[i][j] = (j<i) ? -lr*Bq[i][j] : 0 ----
#pragma unroll
    for (int rep = 0; rep < 16; ++rep) {
      int e = rep * 256 + tid;
      int i = e >> 6, j = e & 63;
      float vv = sA[i * 65 + j];
      sA[i * 65 + j] = (j < i) ? (-TTT_LR) * vv : 0.0f;
    }
    __syncthreads();

    // ---- Y += stril(Bq)*(-lr) * E  (accumulate into sY) ----
    {
      const int trY = (wv >> 1) * 16;
      const int cb = (wv & 1) * 4;
      v8f acc[4];
#pragma unroll
      for (int i = 0; i < 4; ++i) acc[i] = ld_c(sY, 129, trY, (cb + i) * 16, lane);
#pragma unroll
      for (int k0 = 0; k0 < 64; k0 += 4) {
        v2f aa = ld_a(sA, 65, trY, k0, lane);
#pragma unroll
        for (int i = 0; i < 4; ++i)
          acc[i] = WMMA_F32(aa, ld_b(sE, 129, k0, (cb + i) * 16, lane), acc[i]);
      }
#pragma unroll
      for (int i = 0; i < 4; ++i) st_c(sY, 129, trY, (cb + i) * 16, lane, acc[i]);
    }
    __syncthreads();

    // ---- out = x + Y ----
#pragma unroll
    for (int rep = 0; rep < 32; ++rep) {
      int e = rep * 256 + tid;
      int t = e >> 7, d = e & 127;
      go[(size_t)t * DD + d] = gx[(size_t)t * DD + d] + sY[t * 129 + d];
    }

    // ---- W -= lr * E^T K   (A-frag = -lr*E[t][d] transposed on the fly) ----
    {
      const int m0 = wv * 16;  // d-rows of W tile
      v8f w[4];
#pragma unroll
      for (int i = 0; i < 4; ++i) w[i] = ld_c(sW, 65, m0, i * 16, lane);
      const int mA = m0 + (lane & 15);
#pragma unroll
      for (int k0 = 0; k0 < 64; k0 += 4) {
        const int kA = k0 + ((lane >> 4) << 1);
        v2f aa;
        aa.x = -TTT_LR * sE[kA * 129 + mA];
        aa.y = -TTT_LR * sE[(kA + 1) * 129 + mA];
#pragma unroll
        for (int i = 0; i < 4; ++i)
          w[i] = WMMA_F32(aa, ld_b(sK, 65, k0, i * 16, lane), w[i]);
      }
#pragma unroll
      for (int i = 0; i < 4; ++i) st_c(sW, 65, m0, i * 16, lane, w[i]);
    }
    __syncthreads();
  }
}

// ---------------- host-side launch ------------------------------------------
extern "C" void kernel_launch(void* const* d_in, const int* in_sizes, int n_in,
                              void* d_out, int out_size, void* d_ws, size_t ws_size,
                              hipStream_t stream) {
  const float* x     = (const float*)d_in[0];
  const float* Wk    = (const float*)d_in[1];
  const float* bk    = (const float*)d_in[2];
  const float* Wv    = (const float*)d_in[3];
  const float* bv    = (const float*)d_in[4];
  const float* Wq    = (const float*)d_in[5];
  const float* bq    = (const float*)d_in[6];
  const float* gamma = (const float*)d_in[7];
  const float* beta  = (const float*)d_in[8];
  float* out = (float*)d_out;

  const size_t BSD = (size_t)BB * SS * DD;     // 8,388,608
  const size_t BSDS = (size_t)BB * SS * DS;    //   524,288
  float* h       = (float*)d_ws;
  float* keys    = h + BSD;
  float* queries = keys + BSDS;
  float* values  = queries + BSDS;

  // 1) LayerNorm
  ln_kernel<<<BB * SS, 256, 0, stream>>>(x, gamma, beta, h);
  // 2) projections (WMMA GEMMs): M = B*S = 8192
  proj_kernel<<<dim3(128, 1), 256, 0, stream>>>(h, Wk, bk, keys, DS);
  proj_kernel<<<dim3(128, 1), 256, 0, stream>>>(h, Wq, bq, queries, DS);
  proj_kernel<<<dim3(128, 16), 256, 0, stream>>>(h, Wv, bv, values, DD);
  // 3) chunked TTT scan, state sharded over (batch, d-tile)
  scan_kernel<<<dim3(BB, DD / 128), 256, 0, stream>>>(x, keys, queries, values, out);
}